// PointNetEncoder_59373627900376
// MI455X (gfx1250) — compile-verified
//
#include <hip/hip_runtime.h>
#include <math.h>

// PointNet++-style encoder on gfx1250.
// All GEMMs mapped to V_WMMA_F32_16X16X4_F32 (exact f32 math vs reference).
// Tile = 16 message rows = the K=16 neighbors of one output point, so the
// WMMA C layout reduction (8 regs + shfl_xor 16) gives max-aggregation free.
// Weights are pre-interleaved into fragment order (row-pair x col x 2) with
// zero padding, so every B fragment is one unconditional global_load_b64.

#define B_   16
#define N_   2048
#define M_   1024
#define K_   16
#define C1_  64
#define C2_  128
#define C3_  256
#define A2S  72   // conv2a LDS row stride (68 cols padded to 72 for b128 stores)

typedef float v2f __attribute__((ext_vector_type(2)));
typedef float v8f __attribute__((ext_vector_type(8)));

static __device__ __forceinline__ v8f v8zero() {
  v8f z = {0.f,0.f,0.f,0.f,0.f,0.f,0.f,0.f};
  return z;
}

static __device__ __forceinline__ v8f wmma4(v2f a, v2f b, v8f c) {
  // D = A(16x4,f32) x B(4x16,f32) + C(16x16,f32)
  return __builtin_amdgcn_wmma_f32_16x16x4_f32(false, a, false, b, (short)0, c,
                                               false, false);
}

static __device__ __forceinline__ float silu_f(float x) {
  return x / (1.0f + expf(-x));
}

// ----------------------------------------------------- weight interleave ---
// Wp[(p*C + c)*2 + {0,1}] = Wsrc[2p][c], Wsrc[2p+1][c]  (zero past real rows)
__global__ void __launch_bounds__(256) k_prep_w(
    const float* __restrict__ W1, const float* __restrict__ W2a,
    const float* __restrict__ W2b,
    float* __restrict__ W1p, float* __restrict__ W2ap, float* __restrict__ W2bp) {
  int t = blockIdx.x * 256 + threadIdx.x;
  if (t < 4 * C1_) {                       // 8 padded rows -> 4 pairs x 64
    const int p = t >> 6, c = t & 63;
    const int r0 = 2 * p, r1 = 2 * p + 1;
    W1p[t * 2 + 0] = (r0 < 6) ? W1[r0 * C1_ + c] : 0.f;
    W1p[t * 2 + 1] = (r1 < 6) ? W1[r1 * C1_ + c] : 0.f;
    return;
  }
  t -= 4 * C1_;
  if (t < 34 * C2_) {                      // 68 padded rows -> 34 pairs x 128
    const int p = t >> 7, c = t & 127;
    const int r0 = 2 * p, r1 = 2 * p + 1;
    W2ap[t * 2 + 0] = (r0 < 67) ? W2a[r0 * C2_ + c] : 0.f;
    W2ap[t * 2 + 1] = (r1 < 67) ? W2a[r1 * C2_ + c] : 0.f;
    return;
  }
  t -= 34 * C2_;
  if (t < 64 * C3_) {                      // 128 rows -> 64 pairs x 256
    const int p = t >> 8, c = t & 255;
    W2bp[t * 2 + 0] = W2b[(2 * p) * C3_ + c];
    W2bp[t * 2 + 1] = W2b[(2 * p + 1) * C3_ + c];
  }
}

// ---------------------------------------------------------------- kNN ------
__global__ void __launch_bounds__(256) k_knn(const float* __restrict__ pos,
                                             int N, int* __restrict__ nbr) {
  const int b = blockIdx.y;
  const int q = blockIdx.x * 256 + threadIdx.x;
  const float* pb = pos + (size_t)b * N * 3;
  const float qx = pb[q * 3 + 0], qy = pb[q * 3 + 1], qz = pb[q * 3 + 2];

  float bd[K_];
  int   bi[K_];
#pragma unroll
  for (int j = 0; j < K_; ++j) { bd[j] = 3.4e38f; bi[j] = 0; }

  __shared__ float cx[256], cy[256], cz[256];
  for (int t0 = 0; t0 < N; t0 += 256) {
    const int c = t0 + threadIdx.x;
    cx[threadIdx.x] = pb[c * 3 + 0];
    cy[threadIdx.x] = pb[c * 3 + 1];
    cz[threadIdx.x] = pb[c * 3 + 2];
    __syncthreads();
    for (int t = 0; t < 256; ++t) {
      const float dx = qx - cx[t], dy = qy - cy[t], dz = qz - cz[t];
      const float d = dx * dx + dy * dy + dz * dz;
      if (d < bd[K_ - 1]) {
        bd[K_ - 1] = d; bi[K_ - 1] = t0 + t;
#pragma unroll
        for (int j = K_ - 1; j > 0; --j) {
          if (bd[j] < bd[j - 1]) {
            float td = bd[j]; bd[j] = bd[j - 1]; bd[j - 1] = td;
            int   ti = bi[j]; bi[j] = bi[j - 1]; bi[j - 1] = ti;
          }
        }
      }
    }
    __syncthreads();
  }
  int* o = nbr + ((size_t)b * N + q) * K_;
#pragma unroll
  for (int j = 0; j < K_; ++j) o[j] = bi[j];
}

// ------------------------------------------------------------- conv1 -------
// y[16,64] = msg[16,8(pad)] @ W1pad[8,64] + b1, via 2 k-steps x 4 n-tiles.
static __device__ __forceinline__ void conv1_accum(
    const float* __restrict__ pos, const int* __restrict__ idx1,
    const v2f* __restrict__ W1p, const float* __restrict__ b1,
    int b, int n, float* alds /*16x8*/, int lane, v8f acc[4]) {
  const float* pb = pos + (size_t)b * N_ * 3;
  const int*   nb = idx1 + ((size_t)b * N_ + n) * K_;
  const int row = lane & 15, half = lane >> 4;
  const int j = nb[row];
  const float pix = pb[n * 3 + 0], piy = pb[n * 3 + 1], piz = pb[n * 3 + 2];
  const float pjx = pb[j * 3 + 0], pjy = pb[j * 3 + 1], pjz = pb[j * 3 + 2];
  float v0, v1, v2, v3;
  if (half == 0) { v0 = pjx; v1 = pjy; v2 = pjz; v3 = pjx - pix; }
  else           { v0 = pjy - piy; v1 = pjz - piz; v2 = 0.f; v3 = 0.f; }
  float* dst = alds + row * 8 + half * 4;
  dst[0] = v0; dst[1] = v1; dst[2] = v2; dst[3] = v3;

#pragma unroll
  for (int nt = 0; nt < 4; ++nt) acc[nt] = v8zero();
#pragma unroll
  for (int ks = 0; ks < 2; ++ks) {
    const float* ap = alds + row * 8 + ks * 4 + half * 2;
    v2f a; a.x = ap[0]; a.y = ap[1];
    const int base = (ks * 2 + half) * C1_ + row;
    v2f bb[4];
#pragma unroll
    for (int nt = 0; nt < 4; ++nt) bb[nt] = W1p[base + nt * 16];
#pragma unroll
    for (int nt = 0; nt < 4; ++nt) acc[nt] = wmma4(a, bb[nt], acc[nt]);
  }
#pragma unroll
  for (int nt = 0; nt < 4; ++nt) acc[nt] += b1[nt * 16 + row];
}

__global__ void __launch_bounds__(256) k_conv1_stats(
    const float* __restrict__ pos, const int* __restrict__ idx1,
    const v2f* __restrict__ W1p, const float* __restrict__ b1,
    float* __restrict__ gsum, float* __restrict__ gsq) {
  __shared__ float alds[8 * 16 * 8];
  __shared__ float ssum[C1_], ssq[C1_];
  const int tid = threadIdx.x;
  if (tid < C1_) { ssum[tid] = 0.f; ssq[tid] = 0.f; }
  __syncthreads();
  const int wave = tid >> 5, lane = tid & 31;
  const int tile = blockIdx.x * 8 + wave;       // tile = b*N + n
  const int b = tile >> 11, n = tile & (N_ - 1);
  v8f acc[4];
  conv1_accum(pos, idx1, W1p, b1, b, n, alds + wave * 128, lane, acc);
  const int col = lane & 15;
#pragma unroll
  for (int nt = 0; nt < 4; ++nt) {
    float s = 0.f, q = 0.f;
#pragma unroll
    for (int i = 0; i < 8; ++i) { const float v = acc[nt][i]; s += v; q += v * v; }
    s += __shfl_xor(s, 16, 32);
    q += __shfl_xor(q, 16, 32);
    if (lane < 16) {
      atomicAdd(&ssum[nt * 16 + col], s);
      atomicAdd(&ssq[nt * 16 + col], q);
    }
  }
  __syncthreads();
  if (tid < C1_) { atomicAdd(&gsum[tid], ssum[tid]); atomicAdd(&gsq[tid], ssq[tid]); }
}

__global__ void __launch_bounds__(256) k_conv1_apply(
    const float* __restrict__ pos, const int* __restrict__ idx1,
    const v2f* __restrict__ W1p, const float* __restrict__ b1,
    const float* __restrict__ mu, const float* __restrict__ inv,
    const float* __restrict__ g, const float* __restrict__ be,
    float* __restrict__ h1) {
  __shared__ float alds[8 * 16 * 8];
  const int tid = threadIdx.x;
  const int wave = tid >> 5, lane = tid & 31;
  const int tile = blockIdx.x * 8 + wave;
  const int b = tile >> 11, n = tile & (N_ - 1);
  v8f acc[4];
  conv1_accum(pos, idx1, W1p, b1, b, n, alds + wave * 128, lane, acc);
  const int col = lane & 15;
#pragma unroll
  for (int nt = 0; nt < 4; ++nt) {
    const int c = nt * 16 + col;
    const float m = mu[c], iv = inv[c], ga = g[c], bb = be[c];
    float mx = -3.4e38f;
#pragma unroll
    for (int i = 0; i < 8; ++i) {
      const float yb = (acc[nt][i] - m) * iv * ga + bb;
      mx = fmaxf(mx, silu_f(yb));
    }
    mx = fmaxf(mx, __shfl_xor(mx, 16, 32));
    if (lane < 16) h1[(size_t)tile * C1_ + c] = mx;
  }
}

// --------------------------------------------------------------- FPS -------
// One workgroup per cloud; argmax via wave32 shfl reduction (3 barriers/iter).
__global__ void __launch_bounds__(256) k_fps(
    const float* __restrict__ pos, const float* __restrict__ h1,
    int* __restrict__ idxf, float* __restrict__ pos2, float* __restrict__ hs) {
  const int b = blockIdx.x, tid = threadIdx.x;
  const int wave = tid >> 5, lane = tid & 31;
  __shared__ float mind[N_];
  __shared__ float rval[8];
  __shared__ int   ridx[8];
  const float* pb = pos + (size_t)b * N_ * 3;
  const float p0x = pb[0], p0y = pb[1], p0z = pb[2];
  for (int n = tid; n < N_; n += 256) {
    const float dx = pb[n * 3 + 0] - p0x;
    const float dy = pb[n * 3 + 1] - p0y;
    const float dz = pb[n * 3 + 2] - p0z;
    mind[n] = dx * dx + dy * dy + dz * dz;
  }
  if (tid == 0) {
    idxf[b * M_] = 0;
    pos2[(size_t)(b * M_) * 3 + 0] = p0x;
    pos2[(size_t)(b * M_) * 3 + 1] = p0y;
    pos2[(size_t)(b * M_) * 3 + 2] = p0z;
    ridx[0] = 0;
  }
  if (tid < C1_) hs[(size_t)(b * M_) * C1_ + tid] = h1[(size_t)(b * N_) * C1_ + tid];
  __syncthreads();

  for (int s = 1; s < M_; ++s) {
    float bv = -1.f; int bix = 0;
    for (int n = tid; n < N_; n += 256) {
      const float v = mind[n];
      if (v > bv) { bv = v; bix = n; }
    }
    // wave32 shfl argmax (tie -> smaller index)
#pragma unroll
    for (int o = 16; o > 0; o >>= 1) {
      const float ov = __shfl_xor(bv, o, 32);
      const int   oi = __shfl_xor(bix, o, 32);
      if (ov > bv || (ov == bv && oi < bix)) { bv = ov; bix = oi; }
    }
    if (lane == 0) { rval[wave] = bv; ridx[wave] = bix; }
    __syncthreads();
    if (tid == 0) {
      float fb = rval[0]; int fi = ridx[0];
#pragma unroll
      for (int ww = 1; ww < 8; ++ww) {
        if (rval[ww] > fb || (rval[ww] == fb && ridx[ww] < fi)) {
          fb = rval[ww]; fi = ridx[ww];
        }
      }
      ridx[0] = fi;
      idxf[b * M_ + s] = fi;
      pos2[(size_t)(b * M_ + s) * 3 + 0] = pb[fi * 3 + 0];
      pos2[(size_t)(b * M_ + s) * 3 + 1] = pb[fi * 3 + 1];
      pos2[(size_t)(b * M_ + s) * 3 + 2] = pb[fi * 3 + 2];
    }
    __syncthreads();
    const int i = ridx[0];
    const float pix = pb[i * 3 + 0], piy = pb[i * 3 + 1], piz = pb[i * 3 + 2];
    if (tid < C1_)
      hs[(size_t)(b * M_ + s) * C1_ + tid] = h1[((size_t)b * N_ + i) * C1_ + tid];
    for (int n = tid; n < N_; n += 256) {
      const float dx = pb[n * 3 + 0] - pix;
      const float dy = pb[n * 3 + 1] - piy;
      const float dz = pb[n * 3 + 2] - piz;
      const float d = dx * dx + dy * dy + dz * dz;
      if (d < mind[n]) mind[n] = d;
    }
    __syncthreads();
  }
}

// ------------------------------------------------------------- conv2 -------
// y2a[16,128] = msg2[16,68(pad)] @ W2a, 17 k-steps x 8 n-tiles.
// A tile staged with b128 stores (row stride 72 floats keeps 16B alignment).
static __device__ __forceinline__ void conv2a_accum(
    const float* __restrict__ pos2, const float* __restrict__ hs,
    const int* __restrict__ idx2,
    const v2f* __restrict__ W2ap, const float* __restrict__ b2a,
    int b, int m, float* alds /*16xA2S*/, int lane, v8f acc[8]) {
  const float* p2 = pos2 + (size_t)b * M_ * 3;
  const float* hb = hs + (size_t)b * M_ * C1_;
  const int*   nb = idx2 + ((size_t)b * M_ + m) * K_;
  const int row = lane & 15, half = lane >> 4;
  const int j = nb[row];
  const float4* hj4 = (const float4*)(hb + (size_t)j * C1_);
  float4* dst4 = (float4*)(alds + row * A2S);
  if (half == 0) {
#pragma unroll
    for (int q = 0; q < 9; ++q) dst4[q] = hj4[q];          // cols 0..35
  } else {
#pragma unroll
    for (int q = 9; q < 16; ++q) dst4[q] = hj4[q];         // cols 36..63
    dst4[16] = make_float4(p2[j * 3 + 0] - p2[m * 3 + 0],  // cols 64..67
                           p2[j * 3 + 1] - p2[m * 3 + 1],
                           p2[j * 3 + 2] - p2[m * 3 + 2], 0.f);
  }
#pragma unroll
  for (int nt = 0; nt < 8; ++nt) acc[nt] = v8zero();
  for (int ks = 0; ks < 17; ++ks) {
    const float* ap = alds + row * A2S + ks * 4 + half * 2;
    v2f a; a.x = ap[0]; a.y = ap[1];
    const int base = (ks * 2 + half) * C2_ + row;
    v2f bb[8];
#pragma unroll
    for (int nt = 0; nt < 8; ++nt) bb[nt] = W2ap[base + nt * 16];
#pragma unroll
    for (int nt = 0; nt < 8; ++nt) acc[nt] = wmma4(a, bb[nt], acc[nt]);
  }
#pragma unroll
  for (int nt = 0; nt < 8; ++nt) acc[nt] += b2a[nt * 16 + row];
}

// BN+SiLU the 2a accumulators into an LDS 16x128 tile (A for the 2b GEMM)
static __device__ __forceinline__ void conv2_bridge(
    const v8f acc[8], const float* __restrict__ mu, const float* __restrict__ inv,
    const float* __restrict__ g, const float* __restrict__ be,
    float* hl /*16x128*/, int lane) {
  const int col0 = lane & 15, half = lane >> 4;
#pragma unroll
  for (int nt = 0; nt < 8; ++nt) {
    const int c = nt * 16 + col0;
    const float m = mu[c], iv = inv[c], ga = g[c], bb = be[c];
#pragma unroll
    for (int i = 0; i < 8; ++i) {
      const float yb = (acc[nt][i] - m) * iv * ga + bb;
      hl[(i + half * 8) * C2_ + c] = silu_f(yb);
    }
  }
}

// y2b[16,256] = h2a[16,128] @ W2b, 32 k-steps x 16 n-tiles
static __device__ __forceinline__ void conv2b_accum(
    const float* hl, const v2f* __restrict__ W2bp, const float* __restrict__ b2b,
    int lane, v8f acc[16]) {
  const int row = lane & 15, half = lane >> 4;
#pragma unroll
  for (int nt = 0; nt < 16; ++nt) acc[nt] = v8zero();
  for (int ks = 0; ks < 32; ++ks) {
    const float* ap = hl + row * C2_ + ks * 4 + half * 2;
    v2f a; a.x = ap[0]; a.y = ap[1];
    const int base = (ks * 2 + half) * C3_ + row;
    v2f bb[16];
#pragma unroll
    for (int nt = 0; nt < 16; ++nt) bb[nt] = W2bp[base + nt * 16];
#pragma unroll
    for (int nt = 0; nt < 16; ++nt) acc[nt] = wmma4(a, bb[nt], acc[nt]);
  }
#pragma unroll
  for (int nt = 0; nt < 16; ++nt) acc[nt] += b2b[nt * 16 + row];
}

__global__ void __launch_bounds__(128) k_conv2a_stats(
    const float* __restrict__ pos2, const float* __restrict__ hs,
    const int* __restrict__ idx2,
    const v2f* __restrict__ W2ap, const float* __restrict__ b2a,
    float* __restrict__ gsum, float* __restrict__ gsq) {
  __shared__ float alds[4 * 16 * A2S];
  __shared__ float ssum[C2_], ssq[C2_];
  const int tid = threadIdx.x;
  if (tid < C2_) { ssum[tid] = 0.f; ssq[tid] = 0.f; }
  __syncthreads();
  const int wave = tid >> 5, lane = tid & 31;
  const int tile = blockIdx.x * 4 + wave;       // tile = b*M + m
  const int b = tile >> 10, m = tile & (M_ - 1);
  v8f acc[8];
  conv2a_accum(pos2, hs, idx2, W2ap, b2a, b, m, alds + wave * 16 * A2S, lane, acc);
  const int col = lane & 15;
#pragma unroll
  for (int nt = 0; nt < 8; ++nt) {
    float s = 0.f, q = 0.f;
#pragma unroll
    for (int i = 0; i < 8; ++i) { const float v = acc[nt][i]; s += v; q += v * v; }
    s += __shfl_xor(s, 16, 32);
    q += __shfl_xor(q, 16, 32);
    if (lane < 16) {
      atomicAdd(&ssum[nt * 16 + col], s);
      atomicAdd(&ssq[nt * 16 + col], q);
    }
  }
  __syncthreads();
  if (tid < C2_) { atomicAdd(&gsum[tid], ssum[tid]); atomicAdd(&gsq[tid], ssq[tid]); }
}

__global__ void __launch_bounds__(128) k_conv2b_stats(
    const float* __restrict__ pos2, const float* __restrict__ hs,
    const int* __restrict__ idx2,
    const v2f* __restrict__ W2ap, const float* __restrict__ b2a,
    const float* __restrict__ mu2a, const float* __restrict__ inv2a,
    const float* __restrict__ g2a, const float* __restrict__ be2a,
    const v2f* __restrict__ W2bp, const float* __restrict__ b2b,
    float* __restrict__ gsum, float* __restrict__ gsq) {
  __shared__ float alds[4 * 16 * A2S];
  __shared__ float hl[4 * 16 * C2_];
  __shared__ float ssum[C3_], ssq[C3_];
  const int tid = threadIdx.x;
  ssum[tid] = 0.f; ssum[tid + 128] = 0.f;
  ssq[tid] = 0.f;  ssq[tid + 128] = 0.f;
  __syncthreads();
  const int wave = tid >> 5, lane = tid & 31;
  const int tile = blockIdx.x * 4 + wave;
  const int b = tile >> 10, m = tile & (M_ - 1);
  v8f a2a[8];
  conv2a_accum(pos2, hs, idx2, W2ap, b2a, b, m, alds + wave * 16 * A2S, lane, a2a);
  conv2_bridge(a2a, mu2a, inv2a, g2a, be2a, hl + wave * 16 * C2_, lane);
  v8f a2b[16];
  conv2b_accum(hl + wave * 16 * C2_, W2bp, b2b, lane, a2b);
  const int col = lane & 15;
#pragma unroll
  for (int nt = 0; nt < 16; ++nt) {
    float s = 0.f, q = 0.f;
#pragma unroll
    for (int i = 0; i < 8; ++i) { const float v = a2b[nt][i]; s += v; q += v * v; }
    s += __shfl_xor(s, 16, 32);
    q += __shfl_xor(q, 16, 32);
    if (lane < 16) {
      atomicAdd(&ssum[nt * 16 + col], s);
      atomicAdd(&ssq[nt * 16 + col], q);
    }
  }
  __syncthreads();
  atomicAdd(&gsum[tid], ssum[tid]);
  atomicAdd(&gsum[tid + 128], ssum[tid + 128]);
  atomicAdd(&gsq[tid], ssq[tid]);
  atomicAdd(&gsq[tid + 128], ssq[tid + 128]);
}

__global__ void __launch_bounds__(128) k_conv2_apply(
    const float* __restrict__ pos2, const float* __restrict__ hs,
    const int* __restrict__ idx2,
    const v2f* __restrict__ W2ap, const float* __restrict__ b2a,
    const float* __restrict__ mu2a, const float* __restrict__ inv2a,
    const float* __restrict__ g2a, const float* __restrict__ be2a,
    const v2f* __restrict__ W2bp, const float* __restrict__ b2b,
    const float* __restrict__ mu2b, const float* __restrict__ inv2b,
    const float* __restrict__ g2b, const float* __restrict__ be2b,
    float* __restrict__ h2max) {
  __shared__ float alds[4 * 16 * A2S];
  __shared__ float hl[4 * 16 * C2_];
  const int tid = threadIdx.x;
  const int wave = tid >> 5, lane = tid & 31;
  const int tile = blockIdx.x * 4 + wave;
  const int b = tile >> 10, m = tile & (M_ - 1);
  v8f a2a[8];
  conv2a_accum(pos2, hs, idx2, W2ap, b2a, b, m, alds + wave * 16 * A2S, lane, a2a);
  conv2_bridge(a2a, mu2a, inv2a, g2a, be2a, hl + wave * 16 * C2_, lane);
  v8f a2b[16];
  conv2b_accum(hl + wave * 16 * C2_, W2bp, b2b, lane, a2b);
  const int col = lane & 15;
#pragma unroll
  for (int nt = 0; nt < 16; ++nt) {
    const int c = nt * 16 + col;
    const float mm = mu2b[c], iv = inv2b[c], ga = g2b[c], bb = be2b[c];
    float mx = -3.4e38f;
#pragma unroll
    for (int i = 0; i < 8; ++i) {
      const float yb = (a2b[nt][i] - mm) * iv * ga + bb;
      mx = fmaxf(mx, silu_f(yb));
    }
    mx = fmaxf(mx, __shfl_xor(mx, 16, 32));
    if (lane < 16) h2max[(size_t)tile * C3_ + c] = mx;
  }
}

// ----------------------------------------------------------- helpers -------
__global__ void k_bn_finalize(const float* __restrict__ sum,
                              const float* __restrict__ sq,
                              float* __restrict__ mu, float* __restrict__ inv,
                              int C, float invcnt) {
  const int c = blockIdx.x * blockDim.x + threadIdx.x;
  if (c >= C) return;
  const float m = sum[c] * invcnt;
  const float v = sq[c] * invcnt - m * m;
  mu[c] = m;
  inv[c] = rsqrtf(v + 1e-5f);
}

__global__ void __launch_bounds__(256) k_final(const float* __restrict__ h2max,
                                               float* __restrict__ out) {
  const int t = blockIdx.x * 256 + threadIdx.x;   // 16*256 threads
  const int b = t >> 8, c = t & 255;
  const float* p = h2max + (size_t)b * M_ * C3_ + c;
  float mx = -3.4e38f;
  for (int m = 0; m < M_; ++m) mx = fmaxf(mx, p[(size_t)m * C3_]);
  out[t] = mx;
}

// ------------------------------------------------------------- launch ------
extern "C" void kernel_launch(void* const* d_in, const int* in_sizes, int n_in,
                              void* d_out, int out_size, void* d_ws, size_t ws_size,
                              hipStream_t stream) {
  const float* pos  = (const float*)d_in[0];
  const float* W1   = (const float*)d_in[1];
  const float* b1   = (const float*)d_in[2];
  const float* g1   = (const float*)d_in[3];
  const float* be1  = (const float*)d_in[4];
  const float* W2a  = (const float*)d_in[5];
  const float* b2a  = (const float*)d_in[6];
  const float* g2a  = (const float*)d_in[7];
  const float* be2a = (const float*)d_in[8];
  const float* W2b  = (const float*)d_in[9];
  const float* b2b  = (const float*)d_in[10];
  const float* g2b  = (const float*)d_in[11];
  const float* be2b = (const float*)d_in[12];
  float* out = (float*)d_out;

  char* w = (char*)d_ws;
  // stats block: [s1 sum|sq|mu|inv][s2a x4][s2b x4] = 1792 floats
  float* st   = (float*)w;
  float* s1s  = st +    0; float* s1q  = st +   64;
  float* s1m  = st +  128; float* s1i  = st +  192;
  float* s2as = st +  256; float* s2aq = st +  384;
  float* s2am = st +  512; float* s2ai = st +  640;
  float* s2bs = st +  768; float* s2bq = st + 1024;
  float* s2bm = st + 1280; float* s2bi = st + 1536;

  size_t off = 8192;
  float* W1p   = (float*)(w + off); off += (size_t)4  * C1_ * 2 * 4;   // 2 KB
  float* W2ap  = (float*)(w + off); off += (size_t)34 * C2_ * 2 * 4;   // 34 KB
  float* W2bp  = (float*)(w + off); off += (size_t)64 * C3_ * 2 * 4;   // 128 KB
  int*   idx1  = (int*)(w + off);   off += (size_t)B_ * N_ * K_ * 4;   // 2 MB
  int*   idx2  = (int*)(w + off);   off += (size_t)B_ * M_ * K_ * 4;   // 1 MB
  int*   idxf  = (int*)(w + off);   off += (size_t)B_ * M_ * 4;
  float* h1    = (float*)(w + off); off += (size_t)B_ * N_ * C1_ * 4;  // 8 MB
  float* pos2  = (float*)(w + off); off += (size_t)B_ * M_ * 3 * 4;
  float* hs    = (float*)(w + off); off += (size_t)B_ * M_ * C1_ * 4;  // 4 MB
  float* h2max = (float*)(w + off); off += (size_t)B_ * M_ * C3_ * 4;  // 16 MB

  hipMemsetAsync(st, 0, 1792 * sizeof(float), stream);
  k_prep_w<<<82, 256, 0, stream>>>(W1, W2a, W2b, W1p, W2ap, W2bp);

  // conv1 (kNN on raw points, edge MLP, BN stats + apply, max over K)
  k_knn<<<dim3(N_ / 256, B_), 256, 0, stream>>>(pos, N_, idx1);
  k_conv1_stats<<<(B_ * N_) / 8, 256, 0, stream>>>(pos, idx1, (const v2f*)W1p, b1,
                                                   s1s, s1q);
  k_bn_finalize<<<1, C1_, 0, stream>>>(s1s, s1q, s1m, s1i, C1_,
                                       1.0f / (float)(B_ * N_ * K_));
  k_conv1_apply<<<(B_ * N_) / 8, 256, 0, stream>>>(pos, idx1, (const v2f*)W1p, b1,
                                                   s1m, s1i, g1, be1, h1);

  // farthest point sampling + gather of sampled pos/features
  k_fps<<<B_, 256, 0, stream>>>(pos, h1, idxf, pos2, hs);

  // conv2 (kNN on samples, two-layer MLP with per-layer BN stats)
  k_knn<<<dim3(M_ / 256, B_), 256, 0, stream>>>(pos2, M_, idx2);
  k_conv2a_stats<<<(B_ * M_) / 4, 128, 0, stream>>>(pos2, hs, idx2,
                                                    (const v2f*)W2ap, b2a,
                                                    s2as, s2aq);
  k_bn_finalize<<<1, C2_, 0, stream>>>(s2as, s2aq, s2am, s2ai, C2_,
                                       1.0f / (float)(B_ * M_ * K_));
  k_conv2b_stats<<<(B_ * M_) / 4, 128, 0, stream>>>(pos2, hs, idx2,
                                                    (const v2f*)W2ap, b2a,
                                                    s2am, s2ai, g2a, be2a,
                                                    (const v2f*)W2bp, b2b,
                                                    s2bs, s2bq);
  k_bn_finalize<<<1, C3_, 0, stream>>>(s2bs, s2bq, s2bm, s2bi, C3_,
                                       1.0f / (float)(B_ * M_ * K_));
  k_conv2_apply<<<(B_ * M_) / 4, 128, 0, stream>>>(pos2, hs, idx2,
                                                   (const v2f*)W2ap, b2a,
                                                   s2am, s2ai, g2a, be2a,
                                                   (const v2f*)W2bp, b2b,
                                                   s2bm, s2bi, g2b, be2b,
                                                   h2max);

  // global max over the M sampled points
  k_final<<<B_, 256, 0, stream>>>(h2max, out);
  (void)in_sizes; (void)n_in; (void)out_size; (void)ws_size;
}